// CR_549755814203
// MI455X (gfx1250) — compile-verified
//
#include <hip/hip_runtime.h>
#include <hip/hip_bf16.h>

// Controlled-Rx on a 2^23 complex64 state (D=2, control=site0, target=site1).
// Memory-bound: 128 MiB moved, ~34 MFLOP (0.25 FLOP/byte) -> pure b128 streaming,
// ~5.8 us at 23.3 TB/s HBM (faster if L2-resident across replays; 128 MiB < 192 MB L2).
//
// Layout: complex index  idx = control*2^22 + target*2^21 + r
//   control=0 half: out = x (interleave re/im)
//   control=1 half: 2x2 complex rotation between target=0 and target=1 slices
// Output: complex64 -> interleaved float pairs, out[2i]=re, out[2i+1]=im.

__global__ __launch_bounds__(256)
void cr_gate_kernel(const float* __restrict__ xr,
                    const float* __restrict__ xi,
                    const float* __restrict__ angle,
                    float* __restrict__ out,
                    unsigned copyBlocks,   // # blocks handling the identity half
                    unsigned half,         // N/2 (start of control=1 region)
                    unsigned quart)        // N/4 (target stride inside that region)
{
    if (blockIdx.x < copyBlocks) {
        // ---- identity half: interleave copy, 4 complex elements / thread ----
        const unsigned i = (blockIdx.x * 256u + threadIdx.x) * 4u;   // i < half
        const float4 re = *reinterpret_cast<const float4*>(xr + i);
        const float4 im = *reinterpret_cast<const float4*>(xi + i);
        float4* o = reinterpret_cast<float4*>(out + 2u * i);
        o[0] = make_float4(re.x, im.x, re.y, im.y);
        o[1] = make_float4(re.z, im.z, re.w, im.w);
    } else {
        // ---- rotated half: 4 pairs / thread ----
        const unsigned r = ((blockIdx.x - copyBlocks) * 256u + threadIdx.x) * 4u; // r < quart
        // Half-angle is in [0, 2pi]: hardware v_sin/v_cos (via __sincosf) are
        // a few-ulp accurate here and avoid libm's scalarized Payne-Hanek blob.
        const float a = 0.5f * angle[0];
        float s, c;
        __sincosf(a, &s, &c);

        const unsigned i0 = half + r;          // target = 0 slice
        const unsigned i1 = half + quart + r;  // target = 1 slice

        const float4 x0r = *reinterpret_cast<const float4*>(xr + i0);
        const float4 x0i = *reinterpret_cast<const float4*>(xi + i0);
        const float4 x1r = *reinterpret_cast<const float4*>(xr + i1);
        const float4 x1i = *reinterpret_cast<const float4*>(xi + i1);

        // out0 = c*x0 - i*s*x1 ; out1 = c*x1 - i*s*x0
        float4 o0, o1, o2, o3;
        o0.x = fmaf(c, x0r.x,  s * x1i.x);  o0.y = fmaf(c, x0i.x, -s * x1r.x);
        o0.z = fmaf(c, x0r.y,  s * x1i.y);  o0.w = fmaf(c, x0i.y, -s * x1r.y);
        o1.x = fmaf(c, x0r.z,  s * x1i.z);  o1.y = fmaf(c, x0i.z, -s * x1r.z);
        o1.z = fmaf(c, x0r.w,  s * x1i.w);  o1.w = fmaf(c, x0i.w, -s * x1r.w);

        o2.x = fmaf(c, x1r.x,  s * x0i.x);  o2.y = fmaf(c, x1i.x, -s * x0r.x);
        o2.z = fmaf(c, x1r.y,  s * x0i.y);  o2.w = fmaf(c, x1i.y, -s * x0r.y);
        o3.x = fmaf(c, x1r.z,  s * x0i.z);  o3.y = fmaf(c, x1i.z, -s * x0r.z);
        o3.z = fmaf(c, x1r.w,  s * x0i.w);  o3.w = fmaf(c, x1i.w, -s * x0r.w);

        float4* p0 = reinterpret_cast<float4*>(out + 2u * i0);
        p0[0] = o0; p0[1] = o1;
        float4* p1 = reinterpret_cast<float4*>(out + 2u * i1);
        p1[0] = o2; p1[1] = o3;
    }
}

extern "C" void kernel_launch(void* const* d_in, const int* in_sizes, int n_in,
                              void* d_out, int out_size, void* d_ws, size_t ws_size,
                              hipStream_t stream) {
    const float* xr  = (const float*)d_in[0];   // x_real, N floats
    const float* xi  = (const float*)d_in[1];   // x_imag, N floats
    const float* ang = (const float*)d_in[2];   // angle, 1 float
    float* out = (float*)d_out;                 // 2*N floats (complex64 interleaved)

    const unsigned N     = (unsigned)in_sizes[0];   // 2^23
    const unsigned half  = N / 2u;                  // 2^22
    const unsigned quart = N / 4u;                  // 2^21

    const unsigned block       = 256u;
    const unsigned copyThreads = half  / 4u;        // 4 complex elems / thread
    const unsigned rotThreads  = quart / 4u;        // 4 pairs / thread
    const unsigned copyBlocks  = copyThreads / block;              // 4096
    const unsigned totalBlocks = copyBlocks + rotThreads / block;  // 6144

    cr_gate_kernel<<<totalBlocks, block, 0, stream>>>(xr, xi, ang, out,
                                                      copyBlocks, half, quart);
}